// Theta_unfold_29549374997218
// MI455X (gfx1250) — compile-verified
//
#include <hip/hip_runtime.h>
#include <math.h>

#define MFAI 1024
#define MM   (MFAI * MFAI)

typedef __attribute__((ext_vector_type(2))) float v2f;
typedef __attribute__((ext_vector_type(8))) float v8f;

// D = A(16x4 f32) * B(4x16 f32) + C(16x16 f32)
static __device__ inline v8f wmma_acc(v2f a, v2f b, v8f c) {
  // 8-arg pattern: (neg_a, A, neg_b, B, c_mod, C, reuse_a, reuse_b)
  return __builtin_amdgcn_wmma_f32_16x16x4_f32(false, a, false, b, (short)0, c,
                                               false, false);
}

// ---------------- kernel 1: zero the diag-partials workspace -------------
__global__ void init_ws_kernel(float* __restrict__ ws) {
  int t = blockIdx.x * blockDim.x + threadIdx.x;
  if (t < 6 * MFAI) ws[t] = 0.0f;
}

// ---------------- kernel 2: diag of the three complex matmuls ------------
// One wave (32 lanes) per block. Block (kt, jb) accumulates, for k in
// [16*kt, 16*kt+16), the partial sums over j in [32*jb, 32*jb+32) of
//   Qx[k] += A[k,j] * (p[j] * C[j,k])      (complex)
// using V_WMMA_F32_16X16X4_F32 and harvesting only the diagonal.
__global__ __launch_bounds__(32)
void diag_partials_kernel(const float* __restrict__ A1, const float* __restrict__ A2,
                          const float* __restrict__ C1, const float* __restrict__ C2,
                          const float* __restrict__ C3, const float* __restrict__ P1,
                          float* __restrict__ wsd) {
  const int lane = threadIdx.x;
  const int m = lane & 15;   // column within k-tile / row within A-tile
  const int h = lane >> 4;   // half-wave select
  const int k0 = blockIdx.x * 16;
  const int j0b = blockIdx.y * 32;

  v8f q1re = {}, q1im = {}, q2re = {}, q2im = {}, q3re = {}, q3im = {};

  for (int jj = 0; jj < 32; jj += 4) {
    const int j0 = j0b + jj;

    // A-matrix 16x4 f32 layout: VGPR0 = K{0|2}, VGPR1 = K{1|3} -> lane reads
    // contiguous float2 at column j0 + 2*h.
    const size_t aoff = (size_t)(k0 + m) * MFAI + (size_t)(j0 + 2 * h);
    v2f a1_re = *(const v2f*)(A1 + aoff);
    v2f a1_im = *(const v2f*)(A1 + MM + aoff);
    v2f a2_re = *(const v2f*)(A2 + aoff);
    v2f a2_im = *(const v2f*)(A2 + MM + aoff);
    v2f a1_imn = -a1_im;   // fp32 WMMA has no A-negate modifier; negate in VALU
    v2f a2_imn = -a2_im;

    // B-matrix 4x16 f32 layout: VGPR v holds rows {v, v+2} across half-waves.
    const int r0 = j0 + 2 * h;      // row of b[0]
    const int r1 = r0 + 1;          // row of b[1]
    v2f p_re = { P1[r0], P1[r1] };
    v2f p_im = { P1[MFAI + r0], P1[MFAI + r1] };

    const size_t b0 = (size_t)r0 * MFAI + (size_t)(k0 + m);
    const size_t b1 = (size_t)r1 * MFAI + (size_t)(k0 + m);

    v2f c_re, c_im, pc_re, pc_im;

    // ---- Q1 = A1 * (p .* C1) ----
    c_re = (v2f){ C1[b0], C1[b1] };
    c_im = (v2f){ C1[MM + b0], C1[MM + b1] };
    pc_re = p_re * c_re - p_im * c_im;
    pc_im = p_re * c_im + p_im * c_re;
    q1re = wmma_acc(a1_re,  pc_re, q1re);
    q1re = wmma_acc(a1_imn, pc_im, q1re);
    q1im = wmma_acc(a1_re,  pc_im, q1im);
    q1im = wmma_acc(a1_im,  pc_re, q1im);

    // ---- Q2 = A2 * (p .* C2) ----
    c_re = (v2f){ C2[b0], C2[b1] };
    c_im = (v2f){ C2[MM + b0], C2[MM + b1] };
    pc_re = p_re * c_re - p_im * c_im;
    pc_im = p_re * c_im + p_im * c_re;
    q2re = wmma_acc(a2_re,  pc_re, q2re);
    q2re = wmma_acc(a2_imn, pc_im, q2re);
    q2im = wmma_acc(a2_re,  pc_im, q2im);
    q2im = wmma_acc(a2_im,  pc_re, q2im);

    // ---- Q3 = A2 * (p .* C3)  (reference uses A2 here, not A3) ----
    c_re = (v2f){ C3[b0], C3[b1] };
    c_im = (v2f){ C3[MM + b0], C3[MM + b1] };
    pc_re = p_re * c_re - p_im * c_im;
    pc_im = p_re * c_im + p_im * c_re;
    q3re = wmma_acc(a2_re,  pc_re, q3re);
    q3re = wmma_acc(a2_imn, pc_im, q3re);
    q3im = wmma_acc(a2_re,  pc_im, q3im);
    q3im = wmma_acc(a2_im,  pc_re, q3im);
  }

  // Harvest diagonals. C/D layout: VGPR r, lane L -> (M = r + 8*(L/16), N = L%16).
  __shared__ float tile[6][256];
#pragma unroll
  for (int r = 0; r < 8; ++r) {
    const int idx = (r + 8 * h) * 16 + m;
    tile[0][idx] = q1re[r];
    tile[1][idx] = q1im[r];
    tile[2][idx] = q2re[r];
    tile[3][idx] = q2im[r];
    tile[4][idx] = q3re[r];
    tile[5][idx] = q3im[r];
  }
  __syncthreads();
  if (lane < 16) {
#pragma unroll
    for (int q = 0; q < 6; ++q)
      atomicAdd(&wsd[q * MFAI + k0 + lane], tile[q][lane * 16 + lane]);
  }
}

// ---------------- kernel 3: assemble b, normalize, store theta -----------
__global__ void compute_theta_kernel(
    const float* __restrict__ Theta,
    const float* __restrict__ A1, const float* __restrict__ B1, const float* __restrict__ C1,
    const float* __restrict__ A2, const float* __restrict__ B2, const float* __restrict__ C2,
    const float* __restrict__ A3, const float* __restrict__ B3, const float* __restrict__ C3,
    const float* __restrict__ alpha, const float* __restrict__ belta,
    const float* __restrict__ param2, const float* __restrict__ wsd,
    float* __restrict__ theta_out) {
  const int k = blockIdx.x * blockDim.x + threadIdx.x;
  if (k >= MFAI) return;
  const size_t d = (size_t)k * (MFAI + 1);
  const float al = alpha[0], be = belta[0];
  const float th_re = Theta[d], th_im = Theta[MM + d];

  const float q1_re = wsd[0 * MFAI + k], q1_im = wsd[1 * MFAI + k];
  const float q2_re = wsd[2 * MFAI + k], q2_im = wsd[3 * MFAI + k];
  const float q3_re = wsd[4 * MFAI + k], q3_im = wsd[5 * MFAI + k];

  float a_re, a_im, c_re, c_im, x_re, x_im, t_re, t_im;

  // b1 = alpha * dA1*(Thd*dC1) - diag(Q1) + dB1
  a_re = A1[d]; a_im = A1[MM + d]; c_re = C1[d]; c_im = C1[MM + d];
  x_re = th_re * c_re - th_im * c_im;
  x_im = th_re * c_im + th_im * c_re;
  t_re = a_re * x_re - a_im * x_im;
  t_im = a_re * x_im + a_im * x_re;
  float b_re = al * t_re - q1_re + B1[d];
  float b_im = al * t_im - q1_im + B1[MM + d];

  // b2 = belta * dA2*(Thd*dC2) - diag(Q2) + dB2
  a_re = A2[d]; a_im = A2[MM + d]; c_re = C2[d]; c_im = C2[MM + d];
  x_re = th_re * c_re - th_im * c_im;
  x_im = th_re * c_im + th_im * c_re;
  t_re = a_re * x_re - a_im * x_im;
  t_im = a_re * x_im + a_im * x_re;
  b_re += be * t_re - q2_re + B2[d];
  b_im += be * t_im - q2_im + B2[MM + d];

  // b3 = belta * dA3*(Thd*dC3) - diag(Q3) + dB3   (Q3 already built from A2)
  a_re = A3[d]; a_im = A3[MM + d]; c_re = C3[d]; c_im = C3[MM + d];
  x_re = th_re * c_re - th_im * c_im;
  x_im = th_re * c_im + th_im * c_re;
  t_re = a_re * x_re - a_im * x_im;
  t_im = a_re * x_im + a_im * x_re;
  b_re += be * t_re - q3_re + B3[d];
  b_im += be * t_im - q3_im + B3[MM + d];

  b_re += param2[k];
  b_im += param2[MFAI + k];

  const float n = sqrtf(b_re * b_re + b_im * b_im);
  theta_out[k]        = b_re / n;
  theta_out[MFAI + k] = b_im / n;
}

// ---------------- kernel 4: write (2,1024,1024) output (diag matrices) ---
__global__ void write_out_kernel(const float* __restrict__ theta,
                                 float* __restrict__ out) {
  const size_t t = (size_t)blockIdx.x * blockDim.x + threadIdx.x;
  const size_t base = t * 4;              // 4 floats per thread
  const int c = (int)(base >> 20);        // / (1024*1024)
  const int rem = (int)(base & (MM - 1));
  const int i = rem >> 10;
  const int j = rem & 1023;
  const float dg = theta[c * MFAI + i];
  float4 v;
  v.x = (j + 0 == i) ? dg : 0.0f;
  v.y = (j + 1 == i) ? dg : 0.0f;
  v.z = (j + 2 == i) ? dg : 0.0f;
  v.w = (j + 3 == i) ? dg : 0.0f;
  ((float4*)out)[t] = v;
}

extern "C" void kernel_launch(void* const* d_in, const int* in_sizes, int n_in,
                              void* d_out, int out_size, void* d_ws, size_t ws_size,
                              hipStream_t stream) {
  const float* Theta  = (const float*)d_in[0];
  const float* A1     = (const float*)d_in[1];
  const float* B1     = (const float*)d_in[2];
  const float* C1     = (const float*)d_in[3];
  const float* A2     = (const float*)d_in[4];
  const float* B2     = (const float*)d_in[5];
  const float* C2     = (const float*)d_in[6];
  const float* A3     = (const float*)d_in[7];
  const float* B3     = (const float*)d_in[8];
  const float* C3     = (const float*)d_in[9];
  const float* alpha  = (const float*)d_in[10];
  const float* belta  = (const float*)d_in[11];
  const float* param1 = (const float*)d_in[12];
  const float* param2 = (const float*)d_in[13];

  float* out   = (float*)d_out;
  float* wsd   = (float*)d_ws;          // [0, 6*1024): diag partial sums
  float* theta = wsd + 6 * MFAI;        // [6*1024, 8*1024): theta_new (2,1024)

  init_ws_kernel<<<(6 * MFAI + 255) / 256, 256, 0, stream>>>(wsd);

  dim3 grid(MFAI / 16, MFAI / 32);      // 64 k-tiles x 32 j-chunks
  diag_partials_kernel<<<grid, 32, 0, stream>>>(A1, A2, C1, C2, C3, param1, wsd);

  compute_theta_kernel<<<(MFAI + 255) / 256, 256, 0, stream>>>(
      Theta, A1, B1, C1, A2, B2, C2, A3, B3, C3, alpha, belta, param2, wsd,
      theta);

  write_out_kernel<<<(2 * MM / 4) / 256, 256, 0, stream>>>(theta, out);
}